// SpixelAggr_avr_dense_14499809591946
// MI455X (gfx1250) — compile-verified
//
#include <hip/hip_runtime.h>
#include <hip/hip_bf16.h>

// Problem constants (fixed by the reference):
//   B=8, C=128, H=W=256 -> N=65536 pixels/image, S=1024 segments/image
//   input  [B, C, N] f32, seg [B, N] i32, out [B, S, C] f32 (segment means)
#define NPIX      65536
#define NCHAN     128
#define NSEG      1024
#define NIMG      8
#define CG        32      // channels per block (channel group)
#define SUBPX     512     // pixels per TDM sub-tile
#define NSUB      16      // sub-tiles per block  -> 8192 px per block
#define CHUNKPX   (SUBPX * NSUB)

typedef unsigned int v4u __attribute__((ext_vector_type(4)));
typedef int          v4i __attribute__((ext_vector_type(4)));
typedef int          v8i __attribute__((ext_vector_type(8)));

// ---------------- CDNA5 primitives ----------------

// LDS float atomic add (no return) -> ds_add_f32 (DScnt)
__device__ __forceinline__ void ldsAtomicAddF32(float* p, float v) {
  unsigned int off = (unsigned int)(unsigned long long)p;  // flat LDS aperture: low 32 bits = LDS offset
  asm volatile("ds_add_f32 %0, %1" : : "v"(off), "v"(v) : "memory");
}

// Global float atomic add, no return (STOREcnt path) -> global_atomic_add_f32
__device__ __forceinline__ void globalAtomicAddF32(float* p, float v) {
  asm volatile("global_atomic_add_f32 %0, %1, off" : : "v"(p), "v"(v) : "memory");
}

// TDM: DMA a [CG rows x SUBPX cols] f32 tile (row stride NPIX elems) from
// global memory into LDS at lds_off. Issued once per wave; tracked by TENSORcnt.
__device__ __forceinline__ void tdmLoadTile(unsigned int lds_off, const float* gptr) {
  unsigned long long ga = (unsigned long long)gptr;
  v4u g0;
  g0[0] = 1u;                                   // count=1 (valid), user mode, gather off
  g0[1] = lds_off;                              // lds_addr [63:32]
  g0[2] = (unsigned int)ga;                     // global_addr[31:0]
  g0[3] = ((unsigned int)(ga >> 32) & 0x01FFFFFFu) | (2u << 30);  // addr[56:32] | type=2

  v8i g1;
  g1[0] = 0x00020000;                           // workgroup_mask=0, data_size=2 (4 bytes)
  g1[1] = 0;                                    // atomic_barrier_addr=0, tensor_dim0[15:0]=0
  g1[2] = (int)((NPIX >> 16) | (NCHAN << 16));  // tensor_dim0[31:16]=1, tensor_dim1[15:0]=128
  g1[3] = (int)(SUBPX << 16);                   // tensor_dim1[31:16]=0, tile_dim0=512
  g1[4] = CG;                                   // tile_dim1=32, tile_dim2=0
  g1[5] = NPIX;                                 // tensor_dim0_stride[31:0] = 65536
  g1[6] = 0;                                    // stride0 hi | stride1 lo
  g1[7] = 0;

  v4i g2; g2[0] = g2[1] = g2[2] = g2[3] = 0;    // 2D tensor: groups 2/3 unused
  v4i g3; g3[0] = g3[1] = g3[2] = g3[3] = 0;

#if __has_include(<hip/amd_detail/amd_gfx1250_TDM.h>)
  v8i g4; g4[0]=g4[1]=g4[2]=g4[3]=g4[4]=g4[5]=g4[6]=g4[7]=0;
  __builtin_amdgcn_tensor_load_to_lds(g0, g1, g2, g3, g4, 0);
#else
  __builtin_amdgcn_tensor_load_to_lds(g0, g1, g2, g3, 0);
#endif
}

// ---------------- Kernels ----------------

__global__ __launch_bounds__(256) void zero_init_kernel(float* out, unsigned* counts,
                                                        int nOut, int nCnt) {
  int i = blockIdx.x * 256 + threadIdx.x;
  if (i < nOut) out[i] = 0.0f;
  if (i < nCnt) counts[i] = 0u;
}

// Per-image segment pixel counts: LDS histogram then integer global atomics.
__global__ __launch_bounds__(256) void seg_count_kernel(const int* __restrict__ seg,
                                                        unsigned* __restrict__ counts) {
  __shared__ unsigned hist[NSEG];
  const int tid = threadIdx.x;
  const int bid = blockIdx.x;              // 256 blocks, 2048 labels each
  const int b   = bid >> 5;                // 32 blocks per image
  const long long base = (long long)bid * 2048;

  for (int j = tid; j < NSEG; j += 256) hist[j] = 0u;
  __syncthreads();
  for (int i = tid; i < 2048; i += 256) {
    int s = seg[base + i];
    atomicAdd(&hist[s], 1u);               // ds_add_u32
  }
  __syncthreads();
  unsigned* cbase = counts + (long long)b * NSEG;
  for (int j = tid; j < NSEG; j += 256) {
    unsigned h = hist[j];
    if (h) atomicAdd(&cbase[j], h);        // global_atomic_add_u32
  }
}

// Main kernel: TDM-staged, LDS-privatized segment sum.
// Block = (image b, channel group g of 32, pixel chunk of 8192).
// LDS: 128 KB accumulator [1024 seg x 32 ch] (XOR-swizzled) + 2 x 64 KB tile buffers.
__global__ __launch_bounds__(256) void spixel_accum_kernel(const float* __restrict__ in,
                                                           const int* __restrict__ seg,
                                                           float* __restrict__ out) {
  __shared__ float acc[NSEG * CG];         // 128 KB
  __shared__ float tbuf[2][CG * SUBPX];    // 2 x 64 KB

  const int tid   = threadIdx.x;
  const int bid   = blockIdx.x;            // 256 = 8 img * 4 cg * 8 chunks
  const int b     = bid >> 5;
  const int g     = (bid >> 3) & 3;
  const int chunk = bid & 7;

  const long long n0    = (long long)chunk * CHUNKPX;
  const float*    gbase = in  + (((long long)(b * NCHAN + g * CG)) << 16) + n0;
  const int*      sbase = seg + (((long long)b) << 16) + n0;

  for (int i = tid; i < NSEG * CG; i += 256) acc[i] = 0.0f;

  const bool wave0 = (tid < 32);
  if (wave0) tdmLoadTile((unsigned int)(unsigned long long)&tbuf[0][0], gbase);
  __syncthreads();

  for (int t = 0; t < NSUB; ++t) {
    if (t + 1 < NSUB) {
      if (wave0) {
        tdmLoadTile((unsigned int)(unsigned long long)&tbuf[(t + 1) & 1][0],
                    gbase + (long long)(t + 1) * SUBPX);
        __builtin_amdgcn_s_wait_tensorcnt(1);   // tile t landed; t+1 in flight
      }
    } else {
      if (wave0) __builtin_amdgcn_s_wait_tensorcnt(0);
    }
    __syncthreads();

    const float* tp = &tbuf[t & 1][0];          // [CG][SUBPX], px-contiguous
    const int*   sp = sbase + t * SUBPX;
    for (int p = tid; p < SUBPX; p += 256) {
      int s = sp[p];
      unsigned int row = ((unsigned int)s) << 5;
      unsigned int sx  = (unsigned int)s & 31u;
#pragma unroll
      for (int c = 0; c < CG; ++c) {
        // XOR swizzle spreads banks for the random-sid stride-32 rows
        ldsAtomicAddF32(&acc[row | (((unsigned int)c) ^ sx)], tp[c * SUBPX + p]);
      }
    }
    asm volatile("s_wait_dscnt 0" ::: "memory"); // drain asm ds ops before buffer reuse
    __syncthreads();
  }

  // Flush private accumulator with f32 global atomics (8 chunk-blocks contend per row).
  float* obase = out + (((long long)b * NSEG) << 7) + g * CG;
  for (int e = tid; e < NSEG * CG; e += 256) {
    int s = e >> 5;
    int j = e & 31;
    int c = j ^ (s & 31);                        // de-swizzle
    float v = acc[e];
    if (v != 0.0f) globalAtomicAddF32(obase + ((long long)s << 7) + c, v);
  }
}

__global__ __launch_bounds__(256) void finalize_kernel(float* __restrict__ out,
                                                       const unsigned* __restrict__ counts,
                                                       int total) {
  int i = blockIdx.x * 256 + threadIdx.x;
  if (i < total) {
    unsigned c = counts[i >> 7];                 // C = 128
    float d = (c > 0u) ? (float)c : 1.0f;
    out[i] = out[i] / d;
  }
}

// ---------------- Launch ----------------

extern "C" void kernel_launch(void* const* d_in, const int* in_sizes, int n_in,
                              void* d_out, int out_size, void* d_ws, size_t ws_size,
                              hipStream_t stream) {
  const float* in   = (const float*)d_in[0];
  const int*   seg  = (const int*)d_in[1];
  float*       out  = (float*)d_out;                 // [B, S, C] = 1,048,576 f32
  unsigned*    cnts = (unsigned*)d_ws;               // [B*S] = 8192 u32 (32 KB scratch)

  const int nOut = NIMG * NSEG * NCHAN;              // 1048576
  const int nCnt = NIMG * NSEG;                      // 8192

  zero_init_kernel<<<(nOut + 255) / 256, 256, 0, stream>>>(out, cnts, nOut, nCnt);
  seg_count_kernel<<<256, 256, 0, stream>>>(seg, cnts);
  spixel_accum_kernel<<<256, 256, 0, stream>>>(in, seg, out);
  finalize_kernel<<<(nOut + 255) / 256, 256, 0, stream>>>(out, cnts, nOut);
}